// PostProcess_32959579029629
// MI455X (gfx1250) — compile-verified
//
#include <hip/hip_runtime.h>
#include <hip/hip_bf16.h>
#include <float.h>

#define N_ANCH      100800
#define NUM_CLS     80
#define IMG         1280
#define MAX_DETS    300
#define OUT         224
#define SCORE_THRESH 0.1f
#define IOU_THRESH   0.45f

#define SEGLEN 320
#define NSEG   315            /* 315*320 == 100800 */
#define BMWORDS ((N_ANCH + 31) / 32)

#define CROPS_OFF 1800        /* boxes(1200)+scores(300)+classes(300); 16B aligned */
#define KEEP_OFF  (1800 + 301 * 3 * OUT * OUT)   /* 45,310,728 */

/* ---------- helpers ---------- */

__device__ __forceinline__ unsigned ordkey(float f) {
  unsigned u = __float_as_uint(f);
  return (u & 0x80000000u) ? ~u : (u | 0x80000000u);
}
__device__ __forceinline__ float ordinv(unsigned k) {
  unsigned u = (k & 0x80000000u) ? (k ^ 0x80000000u) : ~k;
  return __uint_as_float(u);
}
__device__ __forceinline__ bool allowed_cls(int c) { /* c = argmax + 1 */
  return c == 1 || c == 2 || c == 3 || c == 4 || c == 6 || c == 8 ||
         c == 17 || c == 18 || c == 44;
}

#if defined(__HIP_DEVICE_COMPILE__) && __has_builtin(__builtin_amdgcn_global_load_async_to_lds_b128)
#define HAS_ASYNC_B128 1
#pragma message("CDNA5: async-to-LDS b128 pipeline")
#else
#define HAS_ASYNC_B128 0
#endif
#if defined(__HIP_DEVICE_COMPILE__) && __has_builtin(__builtin_amdgcn_global_load_async_to_lds_b32)
#define HAS_ASYNC_B32 1
#pragma message("CDNA5: async-to-LDS b32 available")
#else
#define HAS_ASYNC_B32 0
#endif

typedef int v4i __attribute__((vector_size(4 * sizeof(int))));
typedef __attribute__((address_space(1))) int*  gptr_t;
typedef __attribute__((address_space(3))) int*  lptr_t;
typedef __attribute__((address_space(1))) v4i*  gptr128_t;
typedef __attribute__((address_space(3))) v4i*  lptr128_t;

/* ---------- kernel 1: per-anchor score / class mask -------------------- */
/* raw layout: channel-major (85, 100800). 4 anchors per thread; 80 logit  */
/* rows streamed through LDS in 10 chunks of 8 rows, double buffered with  */
/* ASYNCcnt-tracked b128 async loads (16B/lane, 512B/wave per instr).      */

#define ANCH_PER_BLK 1024     /* 256 threads * 4 anchors */
#define ROWS_PER_CHUNK 8
#define NCHUNK (NUM_CLS / ROWS_PER_CHUNK)

__global__ __launch_bounds__(256) void k_scores(const float* __restrict__ raw,
                                                float* __restrict__ sc) {
  const int tid = threadIdx.x;
  const int i0  = blockIdx.x * ANCH_PER_BLK + tid * 4;          /* first anchor */
  const int ab  = (i0 <= N_ANCH - 4) ? i0 : (N_ANCH - 4);       /* clamped load base */

  float best[4] = {-FLT_MAX, -FLT_MAX, -FLT_MAX, -FLT_MAX};
  int   bc[4]   = {0, 0, 0, 0};

#if HAS_ASYNC_B128 || HAS_ASYNC_B32
  __shared__ float buf[2][ROWS_PER_CHUNK][ANCH_PER_BLK];

  auto issue = [&](int chunk, int b) {
#pragma unroll
    for (int r = 0; r < ROWS_PER_CHUNK; ++r) {
      const float* g = raw + (size_t)(5 + chunk * ROWS_PER_CHUNK + r) * N_ANCH + ab;
#if HAS_ASYNC_B128
      __builtin_amdgcn_global_load_async_to_lds_b128(
          (gptr128_t)g, (lptr128_t)&buf[b][r][tid * 4], 0, 0);
#else
#pragma unroll
      for (int q = 0; q < 4; ++q)
        __builtin_amdgcn_global_load_async_to_lds_b32(
            (gptr_t)(g + q), (lptr_t)&buf[b][r][tid * 4 + q], 0, 0);
#endif
    }
  };
  auto consume = [&](int chunk, int b) {
#pragma unroll
    for (int r = 0; r < ROWS_PER_CHUNK; ++r) {
      int c = chunk * ROWS_PER_CHUNK + r;
#pragma unroll
      for (int q = 0; q < 4; ++q) {
        float v = buf[b][r][tid * 4 + q];
        if (v > best[q]) { best[q] = v; bc[q] = c; }
      }
    }
  };

  issue(0, 0);
  issue(1, 1);
#pragma unroll
  for (int ch = 0; ch < NCHUNK; ++ch) {
    if (ch < NCHUNK - 1)
      asm volatile("s_wait_asynccnt 8" ::: "memory");   /* chunk ch landed */
    else
      asm volatile("s_wait_asynccnt 0" ::: "memory");
    consume(ch, ch & 1);
    if (ch + 2 < NCHUNK) {
      asm volatile("s_wait_dscnt 0" ::: "memory");      /* buffer reads done */
      issue(ch + 2, ch & 1);
    }
  }
#else
  for (int c = 0; c < NUM_CLS; ++c) {
    const float* g = raw + (size_t)(5 + c) * N_ANCH + ab;
    for (int q = 0; q < 4; ++q) {
      float v = g[q];
      if (v > best[q]) { best[q] = v; bc[q] = c; }
    }
  }
#endif

  /* ab == i0 except in the tail block */
  const int shift = i0 - ab;   /* 0 normally; >0 only for clamped tail lanes */
  if (i0 + 3 < N_ANCH && shift == 0) {
    float4 o;
    float* op = &o.x;
#pragma unroll
    for (int q = 0; q < 4; ++q) {
      int  cls = bc[q] + 1;
      bool m   = (best[q] > SCORE_THRESH) && allowed_cls(cls);
      op[q]    = m ? best[q] : -FLT_MAX;
    }
    *(float4*)(sc + i0) = o;
  } else {
#pragma unroll
    for (int q = 0; q < 4; ++q) {
      int i = ab + q;            /* anchor actually loaded in slot q */
      if (i >= i0 && i < N_ANCH) {
        int  cls = bc[q] + 1;
        bool m   = (best[q] > SCORE_THRESH) && allowed_cls(cls);
        sc[i]    = m ? best[q] : -FLT_MAX;
      }
    }
  }
}

/* ---------- kernel 2: exact top-300 (tournament) ----------------------- */

__global__ __launch_bounds__(1024) void k_topk(const float* __restrict__ sc,
                                               float* __restrict__ topv,
                                               int* __restrict__ topi) {
  __shared__ unsigned long long segkey[NSEG];
  __shared__ unsigned bitmap[BMWORDS];

  const int tid  = threadIdx.x;
  const int lane = tid & 31;
  const int wave = tid >> 5;

  for (int b = tid; b < BMWORDS; b += 1024) bitmap[b] = 0u;

  for (int s = wave; s < NSEG; s += 32) {
    unsigned long long k = 0;
    for (int j = 0; j < SEGLEN / 32; ++j) {
      int   p = s * SEGLEN + j * 32 + lane;
      float v = sc[p];
      unsigned long long key =
          ((unsigned long long)ordkey(v) << 32) | (unsigned)(N_ANCH - p);
      if (key > k) k = key;
    }
#pragma unroll
    for (int off = 16; off; off >>= 1) {
      unsigned long long o = __shfl_xor(k, off, 32);
      if (o > k) k = o;
    }
    if (lane == 0) segkey[s] = k;
  }
  __syncthreads();
  if (wave != 0) return;

  for (int it = 0; it < MAX_DETS; ++it) {
    unsigned long long k = 0;
#pragma unroll
    for (int j = 0; j < 10; ++j) {
      int s = j * 32 + lane;
      if (s < NSEG) {
        unsigned long long v = segkey[s];
        if (v > k) k = v;
      }
    }
#pragma unroll
    for (int off = 16; off; off >>= 1) {
      unsigned long long o = __shfl_xor(k, off, 32);
      if (o > k) k = o;
    }
    int   idx = N_ANCH - (int)(unsigned)(k & 0xffffffffu);
    float val = ordinv((unsigned)(k >> 32));
    if (lane == 0) {
      topv[it] = val;
      topi[it] = idx;
      bitmap[idx >> 5] |= (1u << (idx & 31));
    }
    int s = idx / SEGLEN;
    unsigned long long nk = 0;
    for (int j = 0; j < SEGLEN / 32; ++j) {
      int  p    = s * SEGLEN + j * 32 + lane;
      bool dead = (bitmap[p >> 5] >> (p & 31)) & 1u;
      if (!dead) {
        float v = sc[p];
        unsigned long long key =
            ((unsigned long long)ordkey(v) << 32) | (unsigned)(N_ANCH - p);
        if (key > nk) nk = key;
      }
    }
#pragma unroll
    for (int off = 16; off; off >>= 1) {
      unsigned long long o = __shfl_xor(nk, off, 32);
      if (o > nk) nk = o;
    }
    if (lane == 0) segkey[s] = nk;
  }
}

/* ---------- kernel 3: gather + sequential NMS + scalar heads ----------- */

__global__ __launch_bounds__(512) void k_nms(const float* __restrict__ raw,
                                             const float* __restrict__ topv,
                                             const int* __restrict__ topi,
                                             float* __restrict__ out,
                                             float* __restrict__ wsboxes,
                                             int* __restrict__ wskeep) {
  __shared__ float bxs[MAX_DETS][4];
  __shared__ int   keepS[MAX_DETS];
  __shared__ int   clsS[MAX_DETS];
  __shared__ float valS[MAX_DETS];

  const int t = threadIdx.x;
  if (t < MAX_DETS) {
    int   idx = topi[t];
    float cx  = raw[idx];
    float cy  = raw[(size_t)N_ANCH + idx];
    float w   = raw[(size_t)2 * N_ANCH + idx];
    float h   = raw[(size_t)3 * N_ANCH + idx];
    bxs[t][0] = cx - w * 0.5f;
    bxs[t][1] = cy - h * 0.5f;
    bxs[t][2] = cx + w * 0.5f;
    bxs[t][3] = cy + h * 0.5f;
    float best = -FLT_MAX;
    int   bc   = 0;
    for (int c = 0; c < NUM_CLS; ++c) {
      float v = raw[(size_t)(5 + c) * N_ANCH + idx];
      if (v > best) { best = v; bc = c; }
    }
    clsS[t]  = bc + 1;
    float vv = topv[t];
    valS[t]  = vv;
    keepS[t] = (vv > -1e37f) ? 1 : 0;
  }
  __syncthreads();

  for (int i = 0; i < MAX_DETS - 1; ++i) {
    if (t < MAX_DETS && t > i && keepS[i] && keepS[t]) {
      float ax1 = bxs[i][0], ay1 = bxs[i][1], ax2 = bxs[i][2], ay2 = bxs[i][3];
      float bx1 = bxs[t][0], by1 = bxs[t][1], bx2 = bxs[t][2], by2 = bxs[t][3];
      float areaA = (ax2 - ax1) * (ay2 - ay1);
      float areaB = (bx2 - bx1) * (by2 - by1);
      float ix1 = fmaxf(ax1, bx1), iy1 = fmaxf(ay1, by1);
      float ix2 = fminf(ax2, bx2), iy2 = fminf(ay2, by2);
      float iw = fmaxf(ix2 - ix1, 0.f), ih = fmaxf(iy2 - iy1, 0.f);
      float inter = iw * ih;
      float uni   = fmaxf(areaA + areaB - inter, 1e-9f);
      if (inter / uni > IOU_THRESH) keepS[t] = 0;
    }
    __syncthreads();
  }

  if (t < MAX_DETS) {
    int   kp = keepS[t];
    float b0 = kp ? bxs[t][0] : 0.f;
    float b1 = kp ? bxs[t][1] : 0.f;
    float b2 = kp ? bxs[t][2] : 0.f;
    float b3 = kp ? bxs[t][3] : 0.f;
    out[t * 4 + 0] = b0;
    out[t * 4 + 1] = b1;
    out[t * 4 + 2] = b2;
    out[t * 4 + 3] = b3;
    out[1200 + t]  = kp ? valS[t] : 0.f;
    out[1500 + t]  = kp ? (float)clsS[t] : 0.f;
    out[KEEP_OFF + t] = kp ? 1.f : 0.f;
    wsboxes[t * 4 + 0] = b0;
    wsboxes[t * 4 + 1] = b1;
    wsboxes[t * 4 + 2] = b2;
    wsboxes[t * 4 + 3] = b3;
    wskeep[t] = kp;
  }
}

/* ---------- kernel 4: full-image 1280 -> 224 bilinear resize ----------- */
/* 4 consecutive x-pixels per thread -> one b128 store.                    */

__global__ __launch_bounds__(256) void k_fullcrop(const float* __restrict__ img,
                                                  float* __restrict__ out) {
  const float mean[3]   = {0.485f, 0.456f, 0.406f};
  const float invstd[3] = {1.f / 0.229f, 1.f / 0.224f, 1.f / 0.225f};

  int t4 = blockIdx.x * 256 + threadIdx.x;          /* 0 .. 37631 */
  if (t4 >= 3 * OUT * OUT / 4) return;
  int c  = t4 / (OUT * OUT / 4);
  int r4 = t4 % (OUT * OUT / 4);
  int y  = r4 / (OUT / 4);
  int xq = (r4 % (OUT / 4)) * 4;

  const float s  = (float)IMG / (float)OUT;
  float sy = (y + 0.5f) * s - 0.5f;
  sy = fminf(fmaxf(sy, 0.f), (float)(IMG - 1));
  int   y0 = (int)floorf(sy);
  float ly = sy - y0;
  int   y1 = min(y0 + 1, IMG - 1);

  const float* ch = img + (size_t)c * IMG * IMG;
  float4 o;
  float* op = &o.x;
#pragma unroll
  for (int q = 0; q < 4; ++q) {
    int   x  = xq + q;
    float sx = (x + 0.5f) * s - 0.5f;
    sx = fminf(fmaxf(sx, 0.f), (float)(IMG - 1));
    int   x0 = (int)floorf(sx);
    float lx = sx - x0;
    int   x1 = min(x0 + 1, IMG - 1);
    float v00 = ch[(size_t)y0 * IMG + x0];
    float v01 = ch[(size_t)y0 * IMG + x1];
    float v10 = ch[(size_t)y1 * IMG + x0];
    float v11 = ch[(size_t)y1 * IMG + x1];
    float top = v00 * (1.f - lx) + v01 * lx;
    float bot = v10 * (1.f - lx) + v11 * lx;
    float v   = top * (1.f - ly) + bot * ly;
    op[q] = (v - mean[c]) * invstd[c];
  }
  *(float4*)(out + CROPS_OFF + (size_t)c * OUT * OUT + (size_t)y * OUT + xq) = o;
}

/* ---------- kernel 5: ROI align for the 300 detections ----------------- */
/* 4 consecutive x-pixels per thread -> one b128 store.                    */

__global__ __launch_bounds__(256) void k_roi(const float* __restrict__ img,
                                             const float* __restrict__ boxes,
                                             const int* __restrict__ keep,
                                             float* __restrict__ out) {
  const float mean[3]   = {0.485f, 0.456f, 0.406f};
  const float invstd[3] = {1.f / 0.229f, 1.f / 0.224f, 1.f / 0.225f};

  const int n  = blockIdx.z;
  const int c  = blockIdx.y;
  const int t4 = blockIdx.x * 256 + threadIdx.x;    /* 0 .. 12543 */
  const int y  = t4 / (OUT / 4);
  const int xq = (t4 % (OUT / 4)) * 4;

  float x1 = boxes[n * 4 + 0], y1 = boxes[n * 4 + 1];
  float x2 = boxes[n * 4 + 2], y2 = boxes[n * 4 + 3];
  int   kp = keep[n];

  float gy = (y + 0.5f) / (float)OUT;
  float ys = y1 - 0.5f + gy * (y2 - y1);
  bool  vy = (ys >= -1.0f) && (ys <= (float)IMG);
  float yc = fminf(fmaxf(ys, 0.f), (float)(IMG - 1));
  int   y0 = (int)floorf(yc);
  float ly = yc - y0;
  int   y1i = min(y0 + 1, IMG - 1);

  const float* ch = img + (size_t)c * IMG * IMG;
  float4 o;
  float* op = &o.x;
#pragma unroll
  for (int q = 0; q < 4; ++q) {
    int   x  = xq + q;
    float gx = (x + 0.5f) / (float)OUT;
    float xs = x1 - 0.5f + gx * (x2 - x1);
    bool  vx = (xs >= -1.0f) && (xs <= (float)IMG);
    float xc = fminf(fmaxf(xs, 0.f), (float)(IMG - 1));
    int   x0 = (int)floorf(xc);
    float lx = xc - x0;
    int   x1i = min(x0 + 1, IMG - 1);
    float v = 0.f;
    if (kp && vx && vy) {
      float v00 = ch[(size_t)y0 * IMG + x0];
      float v01 = ch[(size_t)y0 * IMG + x1i];
      float v10 = ch[(size_t)y1i * IMG + x0];
      float v11 = ch[(size_t)y1i * IMG + x1i];
      float top = v00 * (1.f - lx) + v01 * lx;
      float bot = v10 * (1.f - lx) + v11 * lx;
      v = top * (1.f - ly) + bot * ly;
    }
    op[q] = (v - mean[c]) * invstd[c];
  }
  *(float4*)(out + CROPS_OFF + (size_t)((n + 1) * 3 + c) * OUT * OUT +
             (size_t)y * OUT + xq) = o;
}

/* ---------- launcher --------------------------------------------------- */

extern "C" void kernel_launch(void* const* d_in, const int* in_sizes, int n_in,
                              void* d_out, int out_size, void* d_ws, size_t ws_size,
                              hipStream_t stream) {
  (void)in_sizes; (void)n_in; (void)out_size; (void)ws_size;

  const float* raw   = (const float*)d_in[0];   /* (1,85,100800) */
  const float* image = (const float*)d_in[1];   /* (1,3,1280,1280) */
  float* out = (float*)d_out;

  float* ws      = (float*)d_ws;
  float* sc      = ws;                 /* 100800 masked scores      */
  float* topv    = ws + 100800;        /* 300                       */
  int*   topi    = (int*)(ws + 101100);/* 300                       */
  float* wsboxes = ws + 101400;        /* 300*4 final boxes         */
  int*   wskeep  = (int*)(ws + 102600);/* 300 keep flags            */

  k_scores<<<(N_ANCH + ANCH_PER_BLK - 1) / ANCH_PER_BLK, 256, 0, stream>>>(raw, sc);
  k_topk<<<1, 1024, 0, stream>>>(sc, topv, topi);
  k_nms<<<1, 512, 0, stream>>>(raw, topv, topi, out, wsboxes, wskeep);
  k_fullcrop<<<(3 * OUT * OUT / 4 + 255) / 256, 256, 0, stream>>>(image, out);
  dim3 g(OUT * OUT / 4 / 256, 3, MAX_DETS);
  k_roi<<<g, 256, 0, stream>>>(image, wsboxes, wskeep, out);
}